// BasisFunction2D_2293512536823
// MI455X (gfx1250) — compile-verified
//
#include <hip/hip_runtime.h>
#include <hip/hip_bf16.h>
#include <math.h>

// Problem constants (from the reference)
#define NUM_GRIDS 16
#define IN_X 32
#define IN_Z 32
#define OUT 64
#define BATCH 512
#define GP1 (NUM_GRIDS + 1)        // 17 grid borders per axis

typedef __attribute__((ext_vector_type(2))) float v2f;
typedef __attribute__((ext_vector_type(8))) float v8f;

// ---------------------------------------------------------------------------
// Kernel 1: build dense per-axis scatter matrices.
//   Vx[(gx*32 + i)*512 + b] = (gx==idx_x ? 1-dx : gx==idx_x+1 ? dx : 0)
//   Vz[(gz*32 + j)*512 + b] = (gz==idx_z ? 1-dz : gz==idx_z+1 ? dz : 0)
// Each thread owns one (axis, row, batch) column stripe -> no atomics needed.
// ---------------------------------------------------------------------------
__global__ void build_scatter(const float* __restrict__ x,
                              const float* __restrict__ z,
                              const float* __restrict__ borders,
                              const float* __restrict__ invlen,
                              float* __restrict__ Vx,
                              float* __restrict__ Vz) {
  const int per = IN_X * BATCH;                 // 16384 per axis
  int t = blockIdx.x * blockDim.x + threadIdx.x;
  if (t >= 2 * per) return;
  int sel = t / per;                            // 0 -> x axis, 1 -> z axis
  int r   = t - sel * per;
  int i   = r / BATCH;
  int b   = r - i * BATCH;

  float v = sel ? z[i * BATCH + b] : x[i * BATCH + b];
  float* V = sel ? Vz : Vx;

  // Laplace CDF
  float e   = __expf(-fabsf(v));
  float cdf = (v > 0.0f) ? (1.0f - 0.5f * e) : (0.5f * e);
  int idx = (int)(cdf * (float)NUM_GRIDS);
  idx = idx < 0 ? 0 : (idx > NUM_GRIDS - 1 ? NUM_GRIDS - 1 : idx);
  float d = (v - borders[idx]) * invlen[idx];

#pragma unroll
  for (int g = 0; g < GP1; ++g) {
    float w = (g == idx) ? (1.0f - d) : ((g == idx + 1) ? d : 0.0f);
    V[((size_t)g * IN_X + i) * BATCH + b] = w;
  }
}

// ---------------------------------------------------------------------------
// Kernel 2: one wave per (o, 32-batch tile): two 16-col N-tiles share every
// A fragment, halving on-die A traffic and giving 4 independent WMMA chains.
// For each gx (17):
//   C[i, b] = sum_{gz,j} P[gx,gz,o,i,j] * Vz[(gz,j), b]   (dense fp32 WMMA GEMM)
// epilogue: out[o,b] += sum_i Vx[(gx,i), b] * C[i, b]
// A fragment (16x4 f32): lane l holds M=l%16, K = 2*(l/16)+{0,1} (contiguous j)
// B fragment (4x16 f32): lane l holds N=l%16, K = 2*(l/16)+{0,1}
// C fragment: lane l holds N=l%16, rows M = r + 8*(l/16), r=0..7
// ---------------------------------------------------------------------------
__global__ void __launch_bounds__(32) basis_gemm(
    const float* __restrict__ P,    // (17,17,64,32,32)
    const float* __restrict__ Vx,   // (17*32, 512)
    const float* __restrict__ Vz,   // (17*32, 512)
    float* __restrict__ out)        // (64, 512)
{
  const int o    = blockIdx.y;           // 0..63
  const int b0   = blockIdx.x * 32;      // batch tile base (two 16-wide N tiles)
  const int lane = threadIdx.x;          // 0..31 (wave32)
  const int hi   = lane >> 4;            // half-wave select (K pair / row offset)
  const int lo   = lane & 15;            // M (A) / N (B,C) index

  float outacc0 = 0.0f;                  // lanes 0..15: out[o, b0+lo]
  float outacc1 = 0.0f;                  // lanes 0..15: out[o, b0+16+lo]

  const size_t gzStride = (size_t)OUT * IN_X * IN_Z;  // A stride between gz slices

  for (int gx = 0; gx < GP1; ++gx) {
    v8f acc00 = {};   // i = 0..15  half, batch cols b0+0..15
    v8f acc01 = {};   // i = 16..31 half, batch cols b0+0..15
    v8f acc10 = {};   // i = 0..15  half, batch cols b0+16..31
    v8f acc11 = {};   // i = 16..31 half, batch cols b0+16..31

    for (int gz = 0; gz < GP1; ++gz) {
      const float* Ap  = P + ((((size_t)gx * GP1 + gz) * OUT + o) * IN_X) * IN_Z;
      const float* Bp0 = Vz + (size_t)gz * IN_Z * BATCH + (b0 + lo);
      const float* Bp1 = Bp0 + 16;

      if (gz + 1 < GP1)  // pull next gz A-slice toward L2 (global_prefetch_b8)
        __builtin_prefetch(Ap + gzStride + (size_t)lo * IN_Z, 0, 3);

#pragma unroll
      for (int jj = 0; jj < IN_Z; jj += 4) {
        const int j = jj + 2 * hi;       // this lane's K pair {j, j+1}
        v2f a0 = *(const v2f*)(Ap + (size_t)lo        * IN_Z + j);
        v2f a1 = *(const v2f*)(Ap + (size_t)(16 + lo) * IN_Z + j);
        v2f b0v, b1v;
        b0v.x = Bp0[(size_t)j * BATCH];
        b0v.y = Bp0[(size_t)(j + 1) * BATCH];
        b1v.x = Bp1[(size_t)j * BATCH];
        b1v.y = Bp1[(size_t)(j + 1) * BATCH];
        acc00 = __builtin_amdgcn_wmma_f32_16x16x4_f32(
            false, a0, false, b0v, (short)0, acc00, false, false);
        acc10 = __builtin_amdgcn_wmma_f32_16x16x4_f32(
            false, a0, false, b1v, (short)0, acc10, false, false);
        acc01 = __builtin_amdgcn_wmma_f32_16x16x4_f32(
            false, a1, false, b0v, (short)0, acc01, false, false);
        acc11 = __builtin_amdgcn_wmma_f32_16x16x4_f32(
            false, a1, false, b1v, (short)0, acc11, false, false);
      }
    }

    // Epilogue: weight rows by Vx[gx, i, b] and reduce over all 32 i's,
    // separately for the two batch-column tiles.
    float p0 = 0.0f, p1 = 0.0f;
#pragma unroll
    for (int rr = 0; rr < 8; ++rr) {
      const int m = rr + 8 * hi;                       // row within 16-row tile
      float w00 = Vx[((size_t)gx * IN_X + m)      * BATCH + (b0 + lo)];
      float w01 = Vx[((size_t)gx * IN_X + 16 + m) * BATCH + (b0 + lo)];
      float w10 = Vx[((size_t)gx * IN_X + m)      * BATCH + (b0 + 16 + lo)];
      float w11 = Vx[((size_t)gx * IN_X + 16 + m) * BATCH + (b0 + 16 + lo)];
      p0 += acc00[rr] * w00 + acc01[rr] * w01;
      p1 += acc10[rr] * w10 + acc11[rr] * w11;
    }
    p0 += __shfl_xor(p0, 16, 32);        // combine the two row halves
    p1 += __shfl_xor(p1, 16, 32);
    outacc0 += p0;
    outacc1 += p1;
  }

  if (lane < 16) {
    out[(size_t)o * BATCH + b0 + lane]      = outacc0;
    out[(size_t)o * BATCH + b0 + 16 + lane] = outacc1;
  }
}

// ---------------------------------------------------------------------------
extern "C" void kernel_launch(void* const* d_in, const int* in_sizes, int n_in,
                              void* d_out, int out_size, void* d_ws, size_t ws_size,
                              hipStream_t stream) {
  const float* x       = (const float*)d_in[0];
  const float* z       = (const float*)d_in[1];
  const float* P       = (const float*)d_in[2];
  const float* borders = (const float*)d_in[3];
  const float* invlen  = (const float*)d_in[4];
  float* out = (float*)d_out;

  // Workspace: Vx then Vz, each 17*32*512 floats (~1.06 MB each)
  float* Vx = (float*)d_ws;
  float* Vz = Vx + (size_t)GP1 * IN_X * BATCH;

  {
    int total = 2 * IN_X * BATCH;
    build_scatter<<<dim3((total + 255) / 256), dim3(256), 0, stream>>>(
        x, z, borders, invlen, Vx, Vz);
  }
  basis_gemm<<<dim3(BATCH / 32, OUT), dim3(32), 0, stream>>>(P, Vx, Vz, out);
}